// RGBRenderer_52518860095739
// MI455X (gfx1250) — compile-verified
//
#include <hip/hip_runtime.h>
#include <stdint.h>

// NeRF volume-rendering composite for MI455X (gfx1250).
//
// Roofline: ~320 MiB streamed once @ 23.3 TB/s => ~14 us; ~0.25 GFLOP of math
// -> pure bandwidth problem. One wave32 per ray:
//   - CDNA5 async global->LDS staging (ASYNCcnt) double-buffers the rgb
//     stream (60% of all traffic) behind the scan/exp math
//   - NT hints on the deltas/density stream (320 MiB > 192 MB L2)
//   - wave-wide Hillis-Steele scan gives the exclusive-cumsum transmittance
//   - final 32-lane channel reduction done with V_WMMA_F32_16X16X4_F32
//     against an all-ones B matrix (row-sum trick; B=1 makes its lane layout
//     irrelevant, A/D layouts are the documented ones)

#define NSAMP 256
#define CHUNK 32                 // one wave32 handles 32 samples per step
#define NCHUNK (NSAMP / CHUNK)   // 8
#define WAVES_PER_BLOCK 8

typedef float v2f __attribute__((ext_vector_type(2)));
typedef float v8f __attribute__((ext_vector_type(8)));

// Sum x over all 32 lanes of the wave (result valid in every lane).
// D[m,n] = sum_k A[m,k]*1 with a0=x, a1=0  ->  row m sum = x[m] + x[m+16].
// Summing the 8 D VGPRs gives sum_{m=0..7} in lanes 0-15 and sum_{m=8..15}
// in lanes 16-31; one xor-16 shuffle completes the full reduction.
__device__ __forceinline__ float wave_sum_wmma(float x) {
    v2f a; a[0] = x;    a[1] = 0.0f;
    v2f b; b[0] = 1.0f; b[1] = 1.0f;   // all-ones B: layout-independent
    v8f c = {};
    v8f d = __builtin_amdgcn_wmma_f32_16x16x4_f32(
        /*neg_a=*/false, a, /*neg_b=*/false, b,
        /*c_mod=*/(short)0, c, /*reuse_a=*/false, /*reuse_b=*/false);
    float r = ((d[0] + d[1]) + (d[2] + d[3])) + ((d[4] + d[5]) + (d[6] + d[7]));
    r += __shfl_xor(r, 16, 32);
    return r;
}

__global__ __launch_bounds__(WAVES_PER_BLOCK * 32)
void nerf_composite_kernel(const float* __restrict__ rgb,
                           const float* __restrict__ density,
                           const float* __restrict__ deltas,
                           float* __restrict__ out)
{
    // Double-buffered rgb staging: 8 waves * 2 bufs * 32 samples * 3 floats = 6 KB
    __shared__ float s_rgb[WAVES_PER_BLOCK * 2 * CHUNK * 3];

    const int lane = threadIdx.x & 31;
    const int wave = threadIdx.x >> 5;
    const int ray  = blockIdx.x * WAVES_PER_BLOCK + wave;

    const float* dl = deltas  + (size_t)ray * NSAMP;
    const float* dn = density + (size_t)ray * NSAMP;
    const float* rg = rgb     + (size_t)ray * NSAMP * 3;

    float* s_mine = s_rgb + wave * (2 * CHUNK * 3);
    // Flat addr of an LDS object: low 32 bits = LDS byte offset (ISA 10.2).
    const uint32_t lds_base = (uint32_t)(uintptr_t)s_mine;

    // Async-stage one 32-sample rgb chunk (12 B/lane) into LDS buffer `buf`.
    // 3x b32 per lane (a b128 would over-read 4 B past the array on the last
    // lane of the last chunk). Uses ASYNCcnt, not LOADcnt; the 24-bit
    // inst_offset applies to both the LDS and global addresses.
    auto stage = [&](int chunk, int buf) {
        uint64_t g = (uint64_t)(uintptr_t)(rg + chunk * CHUNK * 3) + (uint32_t)(lane * 12);
        uint32_t l = lds_base + (uint32_t)(buf * CHUNK * 3 * 4) + (uint32_t)(lane * 12);
        asm volatile("global_load_async_to_lds_b32 %0, %1, off"          :: "v"(l), "v"(g) : "memory");
        asm volatile("global_load_async_to_lds_b32 %0, %1, off offset:4" :: "v"(l), "v"(g) : "memory");
        asm volatile("global_load_async_to_lds_b32 %0, %1, off offset:8" :: "v"(l), "v"(g) : "memory");
    };

    stage(0, 0);

    float carry = 0.0f;                       // cumsum carried across chunks
    float acc0 = 0.0f, acc1 = 0.0f, acc2 = 0.0f;

    for (int c = 0; c < NCHUNK; ++c) {
        if (c + 1 < NCHUNK) stage(c + 1, (c + 1) & 1);   // prefetch next chunk

        // delta * density for this lane's sample (single-pass stream -> NT)
        const int s = c * CHUNK + lane;
        const float dd = __builtin_nontemporal_load(dl + s) *
                         __builtin_nontemporal_load(dn + s);

        // Wave-wide inclusive scan (Hillis-Steele, 5 steps on wave32)
        float incl = dd;
        #pragma unroll
        for (int d = 1; d < 32; d <<= 1) {
            float t = __shfl_up(incl, d, 32);
            if (lane >= d) incl += t;
        }
        const float excl  = incl - dd;
        const float alpha = 1.0f - expf(-dd);
        const float trans = expf(-(carry + excl));       // exclusive-cumsum transmittance
        const float w     = alpha * trans;
        carry += __shfl(incl, 31, 32);                   // chunk total -> carry

        // Release this chunk's rgb: async loads retire in order, so <=3
        // outstanding (the next chunk's triple) means ours landed in LDS.
        if (c + 1 < NCHUNK)
            asm volatile("s_wait_asynccnt 0x3" ::: "memory");
        else
            asm volatile("s_wait_asynccnt 0x0" ::: "memory");

        const float* p = s_mine + (c & 1) * (CHUNK * 3) + lane * 3;
        acc0 = fmaf(w, p[0], acc0);
        acc1 = fmaf(w, p[1], acc1);
        acc2 = fmaf(w, p[2], acc2);
    }

    // Cross-lane channel sums via WMMA row-sum reduction (EXEC all-1s here:
    // full blocks, no live divergence). Once per ray -> negligible cost.
    acc0 = wave_sum_wmma(acc0);
    acc1 = wave_sum_wmma(acc1);
    acc2 = wave_sum_wmma(acc2);

    if (lane == 0) {
        float* o = out + (size_t)ray * 3;
        o[0] = acc0;
        o[1] = acc1;
        o[2] = acc2;
    }
}

extern "C" void kernel_launch(void* const* d_in, const int* in_sizes, int n_in,
                              void* d_out, int out_size, void* d_ws, size_t ws_size,
                              hipStream_t stream) {
    const float* rgb     = (const float*)d_in[0];   // [N, 256, 3] f32
    const float* density = (const float*)d_in[1];   // [N, 256, 1] f32
    const float* deltas  = (const float*)d_in[2];   // [N, 256]    f32
    float* out = (float*)d_out;                     // [N, 3]      f32

    const int n_rays = in_sizes[2] / NSAMP;         // 65536
    dim3 grid(n_rays / WAVES_PER_BLOCK);            // 8192 blocks
    dim3 block(WAVES_PER_BLOCK * 32);               // 256 threads = 8 wave32

    nerf_composite_kernel<<<grid, block, 0, stream>>>(rgb, density, deltas, out);
}